// MFoE_temp_27556510171414
// MI455X (gfx1250) — compile-verified
//
#include <hip/hip_runtime.h>
#include <hip/hip_bf16.h>

#define NBF   48      // NB_GROUPS
#define CHN   128     // CHANNELS
#define HW    16384   // 128*128
#define NP    32      // pixels per workgroup
#define NITER 10
#define SCALE_C 0.999f

// ws layout (float offsets)
#define OFF_PAR  0       // [0]=eps_e [1]=lamb_e [2]=invden
#define OFF_SCAL 16      // scaling[2][48]
#define OFF_QA   128     // Qa[48][4][4]  = SCALE*Q_eff/taus   (row l, col g)
#define OFF_QB   896     // Qb[48][4][4]  = SCALE*Q^T          (row l, col g)
#define OFF_W1   2048    // W1[192][128]  rows r=f*4+g
#define OFF_W2   28672   // W2[128][192]

typedef __attribute__((ext_vector_type(2))) float v2f;
typedef __attribute__((ext_vector_type(8))) float v8f;

__device__ __forceinline__ v8f wmma4(v2f a, v2f b, v8f c) {
  // D = A(16x4 f32) * B(4x16 f32) + C(16x16 f32)
  return __builtin_amdgcn_wmma_f32_16x16x4_f32(false, a, false, b, (short)0, c,
                                               false, false);
}

// Euclidean projection of a 4-vector onto the unit l1 ball
__device__ __forceinline__ void proj4(const float v[4], float r[4]) {
  float a0 = fabsf(v[0]), a1 = fabsf(v[1]), a2 = fabsf(v[2]), a3 = fabsf(v[3]);
  float ssum = a0 + a1 + a2 + a3;
  float s0 = a0, s1 = a1, s2 = a2, s3 = a3, mx, mn;
  mx = fmaxf(s0, s2); mn = fminf(s0, s2); s0 = mx; s2 = mn;
  mx = fmaxf(s1, s3); mn = fminf(s1, s3); s1 = mx; s3 = mn;
  mx = fmaxf(s0, s1); mn = fminf(s0, s1); s0 = mx; s1 = mn;
  mx = fmaxf(s2, s3); mn = fminf(s2, s3); s2 = mx; s3 = mn;
  mx = fmaxf(s1, s2); mn = fminf(s1, s2); s1 = mx; s2 = mn;
  float c1 = s0 - 1.f, c2 = c1 + s1, c3 = c2 + s2, c4 = c3 + s3;
  float theta = c1;                                   // cond1 always true
  theta = (s1 * 2.f > c2) ? c2 * 0.5f        : theta;
  theta = (s2 * 3.f > c3) ? c3 * (1.f/3.f)   : theta;
  theta = (s3 * 4.f > c4) ? c4 * 0.25f       : theta;
  theta = fmaxf(theta, 0.f);
  float p0 = copysignf(fmaxf(a0 - theta, 0.f), v[0]);
  float p1 = copysignf(fmaxf(a1 - theta, 0.f), v[1]);
  float p2 = copysignf(fmaxf(a2 - theta, 0.f), v[2]);
  float p3 = copysignf(fmaxf(a3 - theta, 0.f), v[3]);
  bool inside = (ssum <= 1.f);
  r[0] = inside ? v[0] : p0;
  r[1] = inside ? v[1] : p1;
  r[2] = inside ? v[2] : p2;
  r[3] = inside ? v[3] : p3;
}

// ---------------- setup: fold all iteration-invariant parameters into ws ----
__global__ __launch_bounds__(256) void mfoe_setup(
    const float* __restrict__ sigma, const float* __restrict__ Qp,
    const float* __restrict__ tausp, const float* __restrict__ lambp,
    const float* __restrict__ epsp,  const float* __restrict__ w1,
    const float* __restrict__ b1,    const float* __restrict__ w2,
    const float* __restrict__ b2,    const float* __restrict__ w3,
    const float* __restrict__ b3,    const float* __restrict__ Wf,
    float* __restrict__ ws) {
  __shared__ float ha[NBF], hb[NBF];
  const int t = threadIdx.x;
  const float lamb_e = expf(lambp[0]);
  const float eps_e  = expf(epsp[0]);
  if (t == 0) {
    ws[OFF_PAR + 0] = eps_e;
    ws[OFF_PAR + 1] = lamb_e;
    ws[OFF_PAR + 2] = 1.f / (1.f + lamb_e * (1.f + eps_e));
  }
  if (t < NBF) {
    const int f = t;
    float Q[4][4];
    for (int g = 0; g < 4; ++g) {                 // l1-normalize rows (axis 2)
      float s = 0.f;
      for (int l = 0; l < 4; ++l) s += fabsf(Qp[f * 16 + g * 4 + l]);
      float d = fmaxf(s, 1.f);
      for (int l = 0; l < 4; ++l) Q[g][l] = Qp[f * 16 + g * 4 + l] / d;
    }
    float M[4][4];                                // M = Q^T Q (PSD)
    for (int i = 0; i < 4; ++i)
      for (int j = 0; j < 4; ++j) {
        float s = 0.f;
        for (int g = 0; g < 4; ++g) s += Q[g][i] * Q[g][j];
        M[i][j] = s;
      }
    float v[4] = {0.5f, 0.5f, 0.5f, 0.5f};        // power iter -> lambda_max
    float lam = 0.f;
    for (int it = 0; it < 64; ++it) {
      float wv[4];
      for (int i = 0; i < 4; ++i)
        wv[i] = M[i][0]*v[0] + M[i][1]*v[1] + M[i][2]*v[2] + M[i][3]*v[3];
      float n = sqrtf(wv[0]*wv[0] + wv[1]*wv[1] + wv[2]*wv[2] + wv[3]*wv[3]) + 1e-30f;
      lam = n;
      for (int i = 0; i < 4; ++i) v[i] = wv[i] / n;
    }
    // sigma_max^2 = lam; Q_eff = Q / lam
    float tau = expf(fmaxf(tausp[f], 0.f));
    float ca  = SCALE_C / (lam * tau + 1e-30f);
    for (int l = 0; l < 4; ++l)
      for (int g = 0; g < 4; ++g) {
        ws[OFF_QA + f * 16 + l * 4 + g] = ca * Q[l][g];        // 'flg' einsum
        ws[OFF_QB + f * 16 + l * 4 + g] = SCALE_C * Q[g][l];   // 'fgl' einsum
      }
  }
  // sigma-conditioned MLP -> scaling[b][f]
  for (int b = 0; b < 2; ++b) {
    float sg = sigma[b];
    if (t < NBF) ha[t] = fmaxf((sg * 20.f - 2.f) * w1[t] + b1[t], 0.f);
    __syncthreads();
    if (t < NBF) {
      float a = b2[t];
      for (int k = 0; k < NBF; ++k) a += ha[k] * w2[k * NBF + t];
      hb[t] = fmaxf(a, 0.f);
    }
    __syncthreads();
    if (t < NBF) {
      float a = b3[t];
      for (int k = 0; k < NBF; ++k) a += hb[k] * w3[k * NBF + t];
      ws[OFF_SCAL + b * NBF + t] = fmaxf(a * 0.05f + sg, 0.f) + 1e-9f;
    }
    __syncthreads();
  }
  // pack Wf[g,f,c] into row-major W1[r=f*4+g][c] and its transpose W2[c][r]
  for (int i = t; i < 192 * CHN; i += 256) {
    int r = i >> 7, c = i & 127;
    float val = Wf[((r & 3) * NBF + (r >> 2)) * CHN + c];
    ws[OFF_W1 + i]           = val;
    ws[OFF_W2 + c * 192 + r] = val;
  }
}

// ---------------- one fixed-point iteration (in-place safe on d_out) --------
__global__ __launch_bounds__(256) void mfoe_iter(const float* xsrc,
                                                 const float* __restrict__ xnoisy,
                                                 float* xout,
                                                 const float* __restrict__ ws) {
  __shared__ float Xs[CHN * NP];   // x tile   [128][32]  16KB
  __shared__ float Zs[192 * NP];   // act tile [192][32]  24KB
  __shared__ float sQa[NBF * 16];
  __shared__ float sQb[NBF * 16];
  __shared__ float sScl[NBF];
  __shared__ float sInv[NBF];

  const int tid  = threadIdx.x;
  const int wave = tid >> 5;
  const int lane = tid & 31;
  const int l15  = lane & 15;
  const int lh2  = (lane >> 4) << 1;   // 0 or 2  (K half-select for A/B)
  const int lh8  = (lane >> 4) << 3;   // 0 or 8  (M half-select for C/D)

  const int pixLin = blockIdx.x * NP;
  const int bb   = pixLin >> 14;       // batch
  const int pix0 = pixLin & (HW - 1);
  const size_t gbase = (size_t)bb * CHN * HW + pix0;

  // stage x tile (coalesced 128B rows) + folded tables
  for (int i = tid; i < CHN * NP; i += 256) {
    int c = i >> 5, p = i & (NP - 1);
    Xs[i] = xsrc[gbase + (size_t)c * HW + p];
  }
  for (int i = tid; i < NBF * 16; i += 256) {
    sQa[i] = ws[OFF_QA + i];
    sQb[i] = ws[OFF_QB + i];
  }
  if (tid < NBF) {
    float s = ws[OFF_SCAL + bb * NBF + tid];
    sScl[tid] = s;
    sInv[tid] = 1.f / s;
  }
  const float eps_e  = ws[OFF_PAR + 0];
  const float lamb_e = ws[OFF_PAR + 1];
  const float invden = ws[OFF_PAR + 2];
  __syncthreads();

  // ---- GEMM1: Z = W1[192x128] @ X[128x32], fused per-f nonlinearity ----
  const float* W1 = ws + OFF_W1;
  for (int j = 0; j < 3; ++j) {
    int q = wave * 3 + j;                 // 24 (mt,nt) pairs over 8 waves
    int mt = q >> 1, nt = q & 1;
    int nn = nt * 16 + l15;
    v8f acc = {0.f, 0.f, 0.f, 0.f, 0.f, 0.f, 0.f, 0.f};
    const float* arow = W1 + (mt * 16 + l15) * CHN + lh2;
    #pragma unroll 8
    for (int k = 0; k < CHN; k += 4) {
      v2f a = *(const v2f*)(arow + k);                    // A: M=l15, K=k+lh2..+1
      v2f b;
      b.x = Xs[(k + lh2    ) * NP + nn];                  // B: K=k+lh2+v, N=nn
      b.y = Xs[(k + lh2 + 1) * NP + nn];
      acc = wmma4(a, b, acc);
    }
    float av[8];
    #pragma unroll
    for (int v = 0; v < 8; ++v) av[v] = acc[v];
    // each lane holds rows mbase..mbase+7 = two complete f-groups at pixel nn
    int mbase = mt * 16 + lh8;
    #pragma unroll
    for (int grp = 0; grp < 2; ++grp) {
      int f = (mbase >> 2) + grp;
      float invs = sInv[f], scl = sScl[f];
      float Lx[4], gc[4], y[4], gy[4];
      #pragma unroll
      for (int g = 0; g < 4; ++g) Lx[g] = av[grp * 4 + g] * invs;
      proj4(Lx, gc);
      #pragma unroll
      for (int g = 0; g < 4; ++g) gc[g] += eps_e * Lx[g];       // moreau g
      const float* qa = &sQa[f * 16];
      const float* qb = &sQb[f * 16];
      #pragma unroll
      for (int l = 0; l < 4; ++l)
        y[l] = qa[l*4+0]*Lx[0] + qa[l*4+1]*Lx[1] + qa[l*4+2]*Lx[2] + qa[l*4+3]*Lx[3];
      proj4(y, gy);
      #pragma unroll
      for (int g = 0; g < 4; ++g) gy[g] += eps_e * y[g];
      #pragma unroll
      for (int l = 0; l < 4; ++l) {
        float qg = qb[l*4+0]*gy[0] + qb[l*4+1]*gy[1] + qb[l*4+2]*gy[2] + qb[l*4+3]*gy[3];
        Zs[(mbase + grp * 4 + l) * NP + nn] = lamb_e * (gc[l] - qg) * scl;
      }
    }
  }
  __syncthreads();

  // ---- GEMM2: Y = W2[128x192] @ Z[192x32], fused update ----
  const float* W2 = ws + OFF_W2;
  for (int nt = 0; nt < 2; ++nt) {
    int mt2 = wave;                       // 8 waves cover 8 M-tiles of C=128
    int nn = nt * 16 + l15;
    v8f acc = {0.f, 0.f, 0.f, 0.f, 0.f, 0.f, 0.f, 0.f};
    const float* arow = W2 + (mt2 * 16 + l15) * 192 + lh2;
    #pragma unroll 8
    for (int k = 0; k < 192; k += 4) {
      v2f a = *(const v2f*)(arow + k);
      v2f b;
      b.x = Zs[(k + lh2    ) * NP + nn];
      b.y = Zs[(k + lh2 + 1) * NP + nn];
      acc = wmma4(a, b, acc);
    }
    int cbase = mt2 * 16 + lh8;
    #pragma unroll
    for (int v = 0; v < 8; ++v) {
      int c = cbase + v;
      float xv = Xs[c * NP + nn];
      size_t gi = gbase + (size_t)c * HW + nn;
      float xn = xnoisy[gi];
      float gval = (xv - xn + acc[v]) * invden;
      xout[gi] = xv - gval;               // x_{t+1} = x_t - g
    }
  }
}

extern "C" void kernel_launch(void* const* d_in, const int* in_sizes, int n_in,
                              void* d_out, int out_size, void* d_ws, size_t ws_size,
                              hipStream_t stream) {
  const float* x_noisy = (const float*)d_in[0];
  const float* sigma   = (const float*)d_in[1];
  const float* Qp      = (const float*)d_in[2];
  const float* tausp   = (const float*)d_in[3];
  const float* lamb    = (const float*)d_in[4];
  const float* epso    = (const float*)d_in[5];
  const float* w1      = (const float*)d_in[6];
  const float* b1      = (const float*)d_in[7];
  const float* w2      = (const float*)d_in[8];
  const float* b2      = (const float*)d_in[9];
  const float* w3      = (const float*)d_in[10];
  const float* b3      = (const float*)d_in[11];
  const float* Wf      = (const float*)d_in[12];
  (void)in_sizes; (void)n_in; (void)out_size; (void)ws_size;

  float* ws  = (float*)d_ws;
  float* out = (float*)d_out;

  mfoe_setup<<<1, 256, 0, stream>>>(sigma, Qp, tausp, lamb, epso,
                                    w1, b1, w2, b2, w3, b3, Wf, ws);
  const int nblocks = (2 * HW) / NP;     // 1024 workgroups / iteration
  for (int it = 0; it < NITER; ++it) {
    const float* xsrc = (it == 0) ? x_noisy : (const float*)out;
    mfoe_iter<<<nblocks, 256, 0, stream>>>(xsrc, x_noisy, out, (const float*)ws);
  }
}